// computeInsulation_27522150433236
// MI455X (gfx1250) — compile-verified
//
#include <hip/hip_runtime.h>
#include <hip/hip_bf16.h>

// Problem constants (match reference)
#define BSZ   4
#define HDIM  4096
#define WDIM  4096
#define KWIN  21
#define LINS  4076   // HDIM - KWIN + 1
#define DRV   10
#define LDV   4057   // LINS - 2*DRV + 1
#define LBAND 4055   // LDV - 2
#define EPSF  1e-6f

#define TILES_PER_BATCH 255              // ceil(LINS/16)
#define TOTAL_TILES     (BSZ * TILES_PER_BATCH)

typedef float v2f __attribute__((ext_vector_type(2)));
typedef float v8f __attribute__((ext_vector_type(8)));

#if __has_builtin(__builtin_amdgcn_wmma_f32_16x16x4_f32)
#define HAVE_WMMA_F32 1
#endif

// Banded window matrix W[k][j] = 1 iff j <= k <= j+20
__device__ __forceinline__ float wmask(int k, int j) {
  return (k >= j && k <= j + (KWIN - 1)) ? 1.0f : 0.0f;
}

// Kernel 1: winavg[b][i] = mean of x[b][i:i+21, i:i+21].
// One wave per 16-position diagonal tile. Two-stage banded matmul on the
// f32 WMMA pipe:  H = X(36x36) * W(36x16);  winsum[j] = (W^T * H)[j][j].
__global__ __launch_bounds__(256) void winavg_wmma_kernel(
    const float* __restrict__ x, float* __restrict__ winavg) {
#ifdef HAVE_WMMA_F32
  // Per-wave private H block: 48 rows x 16 cols, rows padded to 17 floats
  // so lane halves (rows r and r+8) land in different LDS banks.
  __shared__ float Hlds[8][48 * 17];
  const int lane = threadIdx.x & 31;
  const int wib  = threadIdx.x >> 5;
  int wg = blockIdx.x * 8 + wib;
  const bool wave_valid = (wg < TOTAL_TILES);
  if (wg >= TOTAL_TILES) wg = TOTAL_TILES - 1;
  const int b  = wg / TILES_PER_BATCH;
  const int t  = wg % TILES_PER_BATCH;
  const int i0 = t * 16;
  const size_t base = (size_t)b * HDIM * WDIM;

  // A-matrix 16x4 f32 layout (ISA 7.12.2): lanes 0-15 hold M=lane with
  // K=0 (vgpr0), K=1 (vgpr1); lanes 16-31 hold K=2, K=3.
  const int nlo   = lane & 15;
  const int khalf = (lane < 16) ? 0 : 2;

  // ---- Stage 1: H[r][j] = sum_{c=j..j+20} X[i0+r][i0+c], r = 0..47
  #pragma unroll
  for (int mc = 0; mc < 3; ++mc) {
    const int r0 = mc * 16;
    const int row = min(i0 + r0 + nlo, HDIM - 1);   // clamp: kills 0*Inf risk
    const float* rowp = x + base + (size_t)row * WDIM;
    v8f acc = {};
    #pragma unroll
    for (int kc = 0; kc < 9; ++kc) {
      const int c0 = kc * 4;
      // i0+c0+khalf is always even; clamp the even base to WDIM-2 so the
      // contiguous [cx, cx+1] pair is always in bounds (clamped columns are
      // multiplied by W=0 for every valid output, so math is unchanged).
      const int cx = min(i0 + c0 + khalf, WDIM - 2);
      v2f a;
      a.x = rowp[cx];
      a.y = rowp[cx + 1];
      v2f bm;                                        // B = W rows c0..c0+3
      bm.x = wmask(c0 + khalf,     nlo);
      bm.y = wmask(c0 + khalf + 1, nlo);
      acc = __builtin_amdgcn_wmma_f32_16x16x4_f32(
          false, a, false, bm, (short)0, acc, false, false);
    }
    // D-layout -> row-major LDS: vgpr v holds M = r0+v (+8 for high lanes)
    const int rbase = r0 + ((lane >> 4) << 3);
    #pragma unroll
    for (int v = 0; v < 8; ++v)
      Hlds[wib][(rbase + v) * 17 + nlo] = acc[v];
  }
  __syncthreads();

  // ---- Stage 2: G = W^T(16x36) * H(36x16); winsum[i0+j] = G[j][j]
  v8f g = {};
  #pragma unroll
  for (int kc = 0; kc < 9; ++kc) {
    const int k0 = kc * 4;
    v2f a2;                                          // A2[m][k] = W[k][m]
    a2.x = wmask(k0 + khalf,     nlo);
    a2.y = wmask(k0 + khalf + 1, nlo);
    v2f b2;                                          // B2[k][n] = H[k][n]
    b2.x = Hlds[wib][(k0 + khalf)     * 17 + nlo];
    b2.y = Hlds[wib][(k0 + khalf + 1) * 17 + nlo];
    g = __builtin_amdgcn_wmma_f32_16x16x4_f32(
        false, a2, false, b2, (short)0, g, false, false);
  }

  // Diagonal extraction from C/D layout: lanes 0..7 hold G[l][l] in g[l];
  // lanes 24..31 hold G[l-16][l-16] in g[l-24].
  const int idx = (lane < 8) ? lane : (lane - 24);
  float dval = g[0];
  #pragma unroll
  for (int v = 1; v < 8; ++v) dval = (idx == v) ? g[v] : dval;
  const bool act = (lane < 8) || (lane >= 24);
  const int j = (lane < 8) ? lane : (lane - 16);
  const int i = i0 + j;
  if (wave_valid && act && i < LINS)
    winavg[b * LINS + i] = dval * (1.0f / (KWIN * KWIN));
#else
  // Scalar fallback: thread-per-position direct 21x21 sum (grid-stride).
  int id = blockIdx.x * blockDim.x + threadIdx.x;
  for (; id < BSZ * LINS; id += gridDim.x * blockDim.x) {
    const int b = id / LINS, i = id % LINS;
    const float* p = x + (size_t)b * HDIM * WDIM + (size_t)i * WDIM + i;
    float s = 0.f;
    for (int r = 0; r < KWIN; ++r) {
      const float* q = p + (size_t)r * WDIM;
      for (int c = 0; c < KWIN; ++c) s += q[c];
    }
    winavg[id] = s * (1.0f / (KWIN * KWIN));
  }
#endif
}

// Kernel 2: per-batch mean of winavg, then iv = log2((winavg+eps)/(mean+eps))
__global__ __launch_bounds__(256) void iv_kernel(
    const float* __restrict__ winavg, float* __restrict__ iv) {
  __shared__ float red[256];
  const int b = blockIdx.x, tid = threadIdx.x;
  float s = 0.f;
  for (int i = tid; i < LINS; i += 256) s += winavg[b * LINS + i];
  red[tid] = s;
  __syncthreads();
  for (int off = 128; off > 0; off >>= 1) {
    if (tid < off) red[tid] += red[tid + off];
    __syncthreads();
  }
  const float mean = red[0] * (1.0f / (float)LINS);
  const float inv  = 1.0f / (mean + EPSF);
  for (int i = tid; i < LINS; i += 256)
    iv[b * LINS + i] = log2f((winavg[b * LINS + i] + EPSF) * inv);
}

// Kernel 3: dv[j] = (sum iv[j+10..j+19] - sum iv[j..j+9]) / 10
__global__ void dv_kernel(const float* __restrict__ iv,
                          float* __restrict__ dv) {
  const int id = blockIdx.x * blockDim.x + threadIdx.x;
  if (id >= BSZ * LDV) return;
  const int b = id / LDV, j = id % LDV;
  const float* p = iv + b * LINS + j;
  float top = 0.f, bot = 0.f;
  #pragma unroll
  for (int m = 0; m < DRV; ++m) { bot += p[m]; top += p[m + DRV]; }
  dv[b * LDV + j] = (top - bot) * (1.0f / DRV);
}

// Kernel 4: band[j] = (dv[j] < 0) && (dv[j+2] > 0), as 0.0/1.0
__global__ void band_kernel(const float* __restrict__ dv,
                            float* __restrict__ band) {
  const int id = blockIdx.x * blockDim.x + threadIdx.x;
  if (id >= BSZ * LBAND) return;
  const int b = id / LBAND, j = id % LBAND;
  const float lv = dv[b * LDV + j];
  const float rv = dv[b * LDV + j + 2];
  band[b * LBAND + j] = (lv < 0.f && rv > 0.f) ? 1.0f : 0.0f;
}

extern "C" void kernel_launch(void* const* d_in, const int* in_sizes, int n_in,
                              void* d_out, int out_size, void* d_ws, size_t ws_size,
                              hipStream_t stream) {
  const float* x = (const float*)d_in[0];      // (4,1,4096,4096) f32
  float* out  = (float*)d_out;
  float* iv   = out;                           // 4*4076
  float* dv   = out + BSZ * LINS;              // 4*4057
  float* band = dv + BSZ * LDV;                // 4*4055
  float* winavg = (float*)d_ws;                // 4*4076 floats scratch

  winavg_wmma_kernel<<<(TOTAL_TILES + 7) / 8, 256, 0, stream>>>(x, winavg);
  iv_kernel<<<BSZ, 256, 0, stream>>>(winavg, iv);
  dv_kernel<<<(BSZ * LDV + 255) / 256, 256, 0, stream>>>(iv, dv);
  band_kernel<<<(BSZ * LBAND + 255) / 256, 256, 0, stream>>>(dv, band);
}